// ECELoss_47588237640108
// MI455X (gfx1250) — compile-verified
//
#include <hip/hip_runtime.h>
#include <hip/hip_bf16.h>
#include <math.h>

// ECE loss, MI455X (gfx1250, wave32).
// Memory-bound: 400MB logits @ 23.3TB/s => ~17us floor. Design streams logits
// exactly once into registers in V_WMMA_F32_16X16X4_F32 A-matrix layout and
// uses WMMA (B = ones) to produce the 16 per-row sum-of-exp values per wave.

typedef __attribute__((ext_vector_type(2))) float v2f;
typedef __attribute__((ext_vector_type(8))) float v8f;

#define NSEG 16          // N_BINS + 1 segments (reference keeps an always-empty last bin)
#define TILE_ROWS 16
#define C4 25            // 100 classes / 4

__device__ __forceinline__ float bperm_f(int srcLane, float v) {
    return __int_as_float(__builtin_amdgcn_ds_bpermute(srcLane << 2, __float_as_int(v)));
}
__device__ __forceinline__ int bperm_i(int srcLane, int v) {
    return __builtin_amdgcn_ds_bpermute(srcLane << 2, v);
}

__global__ void ece_init(float* __restrict__ bins) {
    if (threadIdx.x < 2 * NSEG) bins[threadIdx.x] = 0.0f;
}

__global__ __launch_bounds__(256) void ece_main(const float* __restrict__ logits,
                                                const int* __restrict__ labels,
                                                float* __restrict__ bins,   // ws: [16 conf][16 acc]
                                                int nrows) {
    __shared__ float sConf[NSEG];
    __shared__ float sAcc[NSEG];
    if (threadIdx.x < NSEG) { sConf[threadIdx.x] = 0.0f; sAcc[threadIdx.x] = 0.0f; }
    __syncthreads();

    const int lane  = threadIdx.x & 31;
    const int h     = lane >> 4;       // K-half select of the WMMA A layout
    const int rsub  = lane & 15;       // row within 16-row tile
    const int wavesPerBlock = blockDim.x >> 5;
    const int gwave  = blockIdx.x * wavesPerBlock + (threadIdx.x >> 5);
    const int nwaves = gridDim.x * wavesPerBlock;
    const int ntiles = (nrows + TILE_ROWS - 1) / TILE_ROWS;

    const v2f onesB = {1.0f, 1.0f};    // B = all-ones 4x16 => D[m][n] = sum_k A[m][k] + C

    for (int t = gwave; t < ntiles; t += nwaves) {
        const int r0  = t * TILE_ROWS;
        const int row = r0 + rsub;
        const int rowc = (row < nrows) ? row : (nrows - 1);
        const float* base = logits + (size_t)rowc * 100 + 2 * h;

        // Prime L2 for the next tile this wave will touch (global_prefetch_b8).
        const int tn = t + nwaves;
        if (tn < ntiles) {
            int prow = tn * TILE_ROWS + rsub;
            if (prow >= nrows) prow = nrows - 1;
            const float* nb = logits + (size_t)prow * 100 + 2 * h;
            __builtin_prefetch(nb, 0, 0);
            __builtin_prefetch(nb + 52, 0, 0);
        }

        // Load lane's 25 column-pairs of its row: exactly the f32 16x4 A layout
        // per WMMA step (lanes 0-15: K=4j..4j+1, lanes 16-31: K=4j+2..4j+3).
        v2f d[C4];
#pragma unroll
        for (int j = 0; j < C4; ++j)
            d[j] = *(const v2f*)(base + 4 * j);

        // ---- Pass 1: per-row max + argmax (first-occurrence on ties) ----
        float m = d[0].x;
        int   am = 2 * h;
#pragma unroll
        for (int j = 0; j < C4; ++j) {
            const int c0 = 4 * j + 2 * h;
            const float x0 = d[j].x, x1 = d[j].y;
            if (x0 > m) { m = x0; am = c0; }
            if (x1 > m) { m = x1; am = c0 + 1; }
        }
        // Combine the two half-lanes covering one row (xor-16 bpermute).
        {
            const float mo = bperm_f(lane ^ 16, m);
            const int   ao = bperm_i(lane ^ 16, am);
            if (mo > m || (mo == m && ao < am)) { m = mo; am = ao; }
        }

        // ---- Pass 2: sum of exp(l - max) via chained WMMA accumulation ----
        v8f acc = {0.0f, 0.0f, 0.0f, 0.0f, 0.0f, 0.0f, 0.0f, 0.0f};
#pragma unroll
        for (int j = 0; j < C4; ++j) {
            v2f a;
            a.x = __expf(d[j].x - m);
            a.y = __expf(d[j].y - m);
            acc = __builtin_amdgcn_wmma_f32_16x16x4_f32(
                /*neg_a=*/false, a, /*neg_b=*/false, onesB,
                /*c_mod=*/(short)0, acc, /*reuse_a=*/false, /*reuse_b=*/false);
        }

        // Extract this lane's row-sum: C/D VGPR i holds row i (lanes 0-15)
        // or row i+8 (lanes 16-31). Select acc[rsub&7] then bpermute.
        const int q = rsub & 7;
        const float x0 = (q & 1) ? acc[1] : acc[0];
        const float x1 = (q & 1) ? acc[3] : acc[2];
        const float x2 = (q & 1) ? acc[5] : acc[4];
        const float x3 = (q & 1) ? acc[7] : acc[6];
        const float y0 = (q & 2) ? x1 : x0;
        const float y1 = (q & 2) ? x3 : x2;
        const float z  = (q & 4) ? y1 : y0;
        // row r<8 lives on lane r; row r>=8 lives on lane r+8 (=16+(r-8)).
        const float sumexp = bperm_f((rsub < 8) ? rsub : (rsub + 8), z);

        // ---- Bin + accumulate (lanes 0-15 own one row each) ----
        if (lane < 16 && row < nrows) {
            const float conf = 1.0f / sumexp;            // exp(0)/sum = max softmax prob
            const float aok  = (am == labels[row]) ? 1.0f : 0.0f;
            int bin = (int)ceilf(conf * 15.0f) - 1;      // conf in (k/15,(k+1)/15] -> k
            bin = bin < 0 ? 0 : (bin > 15 ? 15 : bin);
            atomicAdd(&sConf[bin], conf);                // ds_add_f32
            atomicAdd(&sAcc[bin], aok);
        }
    }

    __syncthreads();
    if (threadIdx.x < NSEG) {
        atomicAdd(&bins[threadIdx.x],        sConf[threadIdx.x]);   // global_atomic_add_f32
        atomicAdd(&bins[NSEG + threadIdx.x], sAcc[threadIdx.x]);
    }
}

__global__ void ece_final(const float* __restrict__ bins, float* __restrict__ out, int nrows) {
    const int lane = threadIdx.x & 31;
    float d = 0.0f;
    if (lane < NSEG) d = fabsf(bins[lane] - bins[NSEG + lane]);
#pragma unroll
    for (int off = 8; off >= 1; off >>= 1)
        d += bperm_f(lane ^ off, d);
    if (lane == 0) out[0] = d / (float)nrows;
}

extern "C" void kernel_launch(void* const* d_in, const int* in_sizes, int n_in,
                              void* d_out, int out_size, void* d_ws, size_t ws_size,
                              hipStream_t stream) {
    const float* logits = (const float*)d_in[0];
    const int*   labels = (const int*)d_in[1];
    const int nrows = in_sizes[1];       // classes fixed at 100 per reference
    float* bins = (float*)d_ws;          // 32 floats: conf[16], acc[16]
    float* out  = (float*)d_out;

    ece_init<<<1, 32, 0, stream>>>(bins);
    // 1024 blocks x 8 wave32 = 8192 waves; ~7.6 sixteen-row tiles per wave.
    ece_main<<<1024, 256, 0, stream>>>(logits, labels, bins, nrows);
    ece_final<<<1, 32, 0, stream>>>(bins, out, nrows);
}